// SimpleAttention_49314814493290
// MI455X (gfx1250) — compile-verified
//
#include <hip/hip_runtime.h>
#include <math.h>

#define S_DIM 4096
#define B_DIM 32
#define F_DIM 512
#define H_DIM 512

// fused path: s-chunks of 32 rows per wave
#define CH   32
#define NCH  (S_DIM / CH)      // 128

// fallback two-pass pooling chunking
#define SC    32
#define SCHUNK (S_DIM / SC)    // 128

typedef __attribute__((ext_vector_type(2))) float v2f;
typedef __attribute__((ext_vector_type(8))) float v8f;

// ---------------------------------------------------------------------------
// Phase 0: v[b][f] = sum_h W_key[h][f] * hidden[b][h]   (512x32x512 fp32 GEMM)
// One wave per 16x16 output tile; K-loop of 128 x V_WMMA_F32_16X16X4_F32.
// ---------------------------------------------------------------------------
__global__ void proj_wmma_kernel(const float* __restrict__ W_key,   // [H,F]
                                 const float* __restrict__ hidden,  // [B,H]
                                 float* __restrict__ v_out) {       // [B,F]
    const int lane  = threadIdx.x;         // 0..31, EXEC all ones
    const int m0    = blockIdx.x * 16;     // f tile base
    const int n0    = blockIdx.y * 16;     // b tile base
    const int mn    = lane & 15;
    const int khalf = (lane >> 4) << 1;    // 0 or 2

    v8f acc = {};
    for (int k0 = 0; k0 < H_DIM; k0 += 4) {
        const int k = k0 + khalf;
        v2f af, bf;
        af.x = W_key[(k + 0) * F_DIM + (m0 + mn)];      // A[m][k] = W_key[k][f]
        af.y = W_key[(k + 1) * F_DIM + (m0 + mn)];
        bf.x = hidden[(n0 + mn) * H_DIM + (k + 0)];     // B[k][n] = hidden[b][k]
        bf.y = hidden[(n0 + mn) * H_DIM + (k + 1)];
        acc = __builtin_amdgcn_wmma_f32_16x16x4_f32(
            false, af, false, bf, (short)0, acc, false, false);
    }
    const int mAdd = (lane >> 4) << 3;
#pragma unroll
    for (int r = 0; r < 8; ++r)
        v_out[(n0 + mn) * F_DIM + (m0 + r + mAdd)] = acc[r];
}

// ---------------------------------------------------------------------------
// Phase 0b: c[b] = sum_h b_key[h] * hidden[b][h]
// ---------------------------------------------------------------------------
__global__ void bias_dot_kernel(const float* __restrict__ b_key,
                                const float* __restrict__ hidden,
                                float* __restrict__ c_out) {
    const int b    = threadIdx.x >> 5;
    const int lane = threadIdx.x & 31;
    float acc = 0.f;
    for (int h = lane; h < H_DIM; h += 32)
        acc = fmaf(b_key[h], hidden[b * H_DIM + h], acc);
#pragma unroll
    for (int off = 16; off > 0; off >>= 1)
        acc += __shfl_xor(acc, off, 32);
    if (lane == 0) c_out[b] = acc;
}

// ===========================================================================
// FUSED PATH: single 256 MB stream of `features`.
// Each wave owns (b, 32-row s-chunk): per row, dot with v[b] (shuffle-reduce),
// then online-softmax update of a 512-float accumulator held in registers.
// ===========================================================================
__global__ void fused_score_pool_kernel(const float* __restrict__ features,
                                        const float* __restrict__ vproj,
                                        const float* __restrict__ cbias,
                                        const float* __restrict__ mask,
                                        float* __restrict__ scores,   // [S,B]
                                        float* __restrict__ pacc,     // [NCH,B,F]
                                        float* __restrict__ pstat) {  // [NCH,B,2]
    const int wave = blockIdx.x * (blockDim.x >> 5) + (threadIdx.x >> 5);
    const int lane = threadIdx.x & 31;
    const int b    = wave & (B_DIM - 1);
    const int sc   = wave >> 5;            // 0..NCH-1
    const int s0   = sc * CH;

    const float4* vrow = (const float4*)(vproj + (size_t)b * F_DIM);
    float4 v4[4];
#pragma unroll
    for (int i = 0; i < 4; ++i) v4[i] = vrow[lane + i * 32];

    const float cb    = cbias[b];
    const float mlane = mask[(s0 + lane) * B_DIM + b];  // one mask value/lane (CH==32)

    float4 acc4[4] = {};
    float m = -INFINITY, zsum = 0.f, myscore = 0.f;

    for (int j = 0; j < CH; ++j) {
        const int s = s0 + j;
        const float4* frow =
            (const float4*)(features + ((size_t)s * B_DIM + b) * F_DIM);
        float4 f4[4];
#pragma unroll
        for (int i = 0; i < 4; ++i) f4[i] = frow[lane + i * 32];

        float d = 0.f;
#pragma unroll
        for (int i = 0; i < 4; ++i) {
            d = fmaf(f4[i].x, v4[i].x, d);
            d = fmaf(f4[i].y, v4[i].y, d);
            d = fmaf(f4[i].z, v4[i].z, d);
            d = fmaf(f4[i].w, v4[i].w, d);
        }
#pragma unroll
        for (int off = 16; off > 0; off >>= 1)   // butterfly: all lanes get sum
            d += __shfl_xor(d, off, 32);

        const float score = d + cb + __shfl(mlane, j, 32) * (-99999.0f);
        if (lane == j) myscore = score;          // lane j archives row j's score

        const float newm  = fmaxf(m, score);
        const float scale = __expf(m - newm);    // exp(-inf)=0 on first row
        const float w     = __expf(score - newm);
        zsum = fmaf(zsum, scale, w);
#pragma unroll
        for (int i = 0; i < 4; ++i) {
            acc4[i].x = fmaf(acc4[i].x, scale, w * f4[i].x);
            acc4[i].y = fmaf(acc4[i].y, scale, w * f4[i].y);
            acc4[i].z = fmaf(acc4[i].z, scale, w * f4[i].z);
            acc4[i].w = fmaf(acc4[i].w, scale, w * f4[i].w);
        }
        m = newm;
    }

    scores[(s0 + lane) * B_DIM + b] = myscore;
    float4* po = (float4*)(pacc + ((size_t)sc * B_DIM + b) * F_DIM);
#pragma unroll
    for (int i = 0; i < 4; ++i) po[lane + i * 32] = acc4[i];
    if (lane == 0) {
        pstat[(sc * B_DIM + b) * 2 + 0] = m;
        pstat[(sc * B_DIM + b) * 2 + 1] = zsum;
    }
}

// Merge the NCH chunk-partials per batch with exact log-sum-exp reweighting.
__global__ void merge_pool_kernel(const float* __restrict__ pacc,
                                  const float* __restrict__ pstat,
                                  float* __restrict__ summary,   // [B,F]
                                  float* __restrict__ bstat) {   // [B,2]
    const int b   = blockIdx.x;
    const int tid = threadIdx.x;           // 256
    __shared__ float msh[NCH], ssh[NCH], wsh[NCH];
    if (tid < NCH) {
        msh[tid] = pstat[(tid * B_DIM + b) * 2 + 0];
        ssh[tid] = pstat[(tid * B_DIM + b) * 2 + 1];
    }
    __syncthreads();
    float M = -INFINITY;
    for (int c = 0; c < NCH; ++c) M = fmaxf(M, msh[c]);   // identical in all threads
    if (tid < NCH) wsh[tid] = __expf(msh[tid] - M);
    __syncthreads();
    float Z = 0.f;
    for (int c = 0; c < NCH; ++c) Z = fmaf(wsh[c], ssh[c], Z);
    const float invZ = 1.0f / Z;

    for (int f = tid; f < F_DIM; f += 256) {
        float a = 0.f;
        for (int c = 0; c < NCH; ++c)
            a = fmaf(wsh[c], pacc[((size_t)c * B_DIM + b) * F_DIM + f], a);
        summary[b * F_DIM + f] = a * invZ;
    }
    if (tid == 0) { bstat[b * 2] = M; bstat[b * 2 + 1] = invZ; }
}

// Finalize distribution from archived raw scores.
__global__ void dist_kernel(const float* __restrict__ scores,
                            const float* __restrict__ bstat,
                            float* __restrict__ dist) {
    const int idx = blockIdx.x * 256 + threadIdx.x;  // < S*B, layout [S,B]
    const int b   = idx & (B_DIM - 1);
    dist[idx] = __expf(scores[idx] - bstat[b * 2]) * bstat[b * 2 + 1];
}

// ===========================================================================
// FALLBACK two-pass path (small workspace): scores pass + softmax + pooling.
// ===========================================================================
__global__ void scores_kernel(const float* __restrict__ features,
                              const float* __restrict__ vproj,
                              const float* __restrict__ cbias,
                              const float* __restrict__ mask,
                              float* __restrict__ scores) {
    const int wave = blockIdx.x * (blockDim.x >> 5) + (threadIdx.x >> 5);
    const int lane = threadIdx.x & 31;
    const int b    = wave & (B_DIM - 1);
    const int s    = wave >> 5;
    const float4* frow = (const float4*)(features + ((size_t)s * B_DIM + b) * F_DIM);
    const float4* vrow = (const float4*)(vproj + (size_t)b * F_DIM);
    float acc = 0.f;
#pragma unroll
    for (int i = 0; i < 4; ++i) {
        const float4 f4 = frow[lane + i * 32];
        const float4 v4 = vrow[lane + i * 32];
        acc = fmaf(f4.x, v4.x, acc); acc = fmaf(f4.y, v4.y, acc);
        acc = fmaf(f4.z, v4.z, acc); acc = fmaf(f4.w, v4.w, acc);
    }
#pragma unroll
    for (int off = 16; off > 0; off >>= 1) acc += __shfl_xor(acc, off, 32);
    if (lane == 0)
        scores[s * B_DIM + b] = acc + cbias[b] + mask[s * B_DIM + b] * (-99999.0f);
}

__global__ void softmax_kernel(const float* __restrict__ scores,
                               float* __restrict__ dist) {
    const int b = blockIdx.x, tid = threadIdx.x;
    __shared__ float red[8];
    float vals[16], vmax = -INFINITY;
#pragma unroll
    for (int i = 0; i < 16; ++i) {
        vals[i] = scores[(tid + i * 256) * B_DIM + b];
        vmax = fmaxf(vmax, vals[i]);
    }
#pragma unroll
    for (int off = 16; off > 0; off >>= 1) vmax = fmaxf(vmax, __shfl_xor(vmax, off, 32));
    if ((tid & 31) == 0) red[tid >> 5] = vmax;
    __syncthreads();
    float mm = red[0];
#pragma unroll
    for (int i = 1; i < 8; ++i) mm = fmaxf(mm, red[i]);
    __syncthreads();
    float sum = 0.f;
#pragma unroll
    for (int i = 0; i < 16; ++i) { vals[i] = __expf(vals[i] - mm); sum += vals[i]; }
#pragma unroll
    for (int off = 16; off > 0; off >>= 1) sum += __shfl_xor(sum, off, 32);
    if ((tid & 31) == 0) red[tid >> 5] = sum;
    __syncthreads();
    float tot = 0.f;
#pragma unroll
    for (int i = 0; i < 8; ++i) tot += red[i];
    const float inv = 1.0f / tot;
#pragma unroll
    for (int i = 0; i < 16; ++i) dist[(tid + i * 256) * B_DIM + b] = vals[i] * inv;
}

__global__ void pool_partial_kernel(const float* __restrict__ features,
                                    const float* __restrict__ dist,
                                    float* __restrict__ partial) {
    const int sc = blockIdx.x, b = blockIdx.y;
    const int f  = blockIdx.z * 256 + threadIdx.x;
    float acc = 0.f;
    for (int i = 0; i < SCHUNK; ++i) {
        const int s = sc * SCHUNK + i;
        acc = fmaf(features[((size_t)s * B_DIM + b) * F_DIM + f],
                   dist[s * B_DIM + b], acc);
    }
    partial[((size_t)sc * B_DIM + b) * F_DIM + f] = acc;
}

__global__ void pool_reduce_kernel(const float* __restrict__ partial,
                                   float* __restrict__ summary) {
    const int idx = blockIdx.x * 256 + threadIdx.x;
    float acc = 0.f;
#pragma unroll
    for (int sc = 0; sc < SC; ++sc)
        acc += partial[(size_t)sc * (B_DIM * F_DIM) + idx];
    summary[idx] = acc;
}

// ---------------------------------------------------------------------------
extern "C" void kernel_launch(void* const* d_in, const int* in_sizes, int n_in,
                              void* d_out, int out_size, void* d_ws, size_t ws_size,
                              hipStream_t stream) {
    const float* features = (const float*)d_in[0];  // [S,B,F]
    const float* hidden   = (const float*)d_in[1];  // [1,B,H]
    const float* mask     = (const float*)d_in[2];  // [S,B]
    const float* W_key    = (const float*)d_in[3];  // [H,F]
    const float* b_key    = (const float*)d_in[4];  // [H]

    float* out     = (float*)d_out;
    float* summary = out;                           // 16384 floats
    float* dist    = out + B_DIM * F_DIM;           // 131072 floats

    float* ws    = (float*)d_ws;
    float* vproj = ws;                              // 16384
    float* cbias = ws + 16384;                      // 32 (+pad to 16512)

    proj_wmma_kernel<<<dim3(F_DIM / 16, B_DIM / 16), 32, 0, stream>>>(
        W_key, hidden, vproj);
    bias_dot_kernel<<<1, 1024, 0, stream>>>(b_key, hidden, cbias);

    const size_t fused_floats =
        16512 + (size_t)NCH * B_DIM * 2 + (size_t)S_DIM * B_DIM +
        (size_t)NCH * B_DIM * F_DIM + 2 * B_DIM;    // ~2.25M floats (~9 MB)

    if (ws_size >= fused_floats * sizeof(float)) {
        // ---- fused single-stream path (~12 us roofline) ----
        float* pstat  = ws + 16512;                           // NCH*B*2  = 8192
        float* scores = pstat + NCH * B_DIM * 2;              // S*B      = 131072
        float* pacc   = scores + S_DIM * B_DIM;               // NCH*B*F  = 2097152
        float* bstat  = pacc + (size_t)NCH * B_DIM * F_DIM;   // 64

        fused_score_pool_kernel<<<(NCH * B_DIM) / 8, 256, 0, stream>>>(
            features, vproj, cbias, mask, scores, pacc, pstat);
        merge_pool_kernel<<<B_DIM, 256, 0, stream>>>(pacc, pstat, summary, bstat);
        dist_kernel<<<(S_DIM * B_DIM) / 256, 256, 0, stream>>>(scores, bstat, dist);
    } else {
        // ---- fallback two-stream path (~22 us roofline, ~2.7 MB ws) ----
        float* scores  = ws + 16512;                          // 131072
        float* partial = scores + S_DIM * B_DIM;              // 524288

        scores_kernel<<<(S_DIM * B_DIM) / 8, 256, 0, stream>>>(
            features, vproj, cbias, mask, scores);
        softmax_kernel<<<B_DIM, 256, 0, stream>>>(scores, dist);
        pool_partial_kernel<<<dim3(SC, B_DIM, F_DIM / 256), 256, 0, stream>>>(
            features, dist, partial);
        pool_reduce_kernel<<<(B_DIM * F_DIM) / 256, 256, 0, stream>>>(
            partial, summary);
    }
}